// TTMatrix_47717086659266
// MI455X (gfx1250) — compile-verified
//
#include <hip/hip_runtime.h>

// TT/MPO matvec, fully fused per batch element.
// Stage1 (34.4 GFLOP of the 38.7 total) runs as 16 parallel 16x(K=256)x16
// WMMA-F32 GEMM tiles per o1-slice, with each wave's core1 A-operand held
// persistently in 128 VGPRs (loaded once from L2, reused 16x).
// Dual accumulators per stage break the WMMA D->C dependency chain so
// independent WMMAs can issue back-to-back while LDS loads run ahead.

typedef __attribute__((ext_vector_type(2))) float v2f;
typedef __attribute__((ext_vector_type(8))) float v8f;

#define ST_V 272   // ldsV row stride (floats): conflict-free B-frag reads
#define ST_X 260   // ldsX row stride
#define ST_Y 292   // ldsY per-o2 stride (16 k2-rows of 18 + pad)

__device__ __forceinline__ v8f wmma4(v2f a, v2f b, v8f c) {
  // (neg_a, A, neg_b, B, c_mod, C, reuse_a, reuse_b)
  return __builtin_amdgcn_wmma_f32_16x16x4_f32(false, a, false, b, (short)0, c,
                                               false, false);
}

__global__ __launch_bounds__(512)
void tt_matvec_kernel(const float* __restrict__ vec,
                      const float* __restrict__ core0,
                      const float* __restrict__ core1,
                      const float* __restrict__ core2,
                      float* __restrict__ out) {
  __shared__ float ldsV[16 * ST_V];  // v[b] as [i1][(i2,i3)]
  __shared__ float ldsX[16 * ST_X];  // stage0 out: [k1][(i2,i3)]
  __shared__ float ldsY[16 * ST_Y];  // stage1 out: [o2][k2(x18)][i3]
  __shared__ float ldsT[256];        // stage2 out tile [o3][o2]

  const int tid  = threadIdx.x;
  const int wave = tid >> 5;
  const int lane = tid & 31;
  const int half = lane >> 4;  // WMMA half-wave (K split)
  const int l16  = lane & 15;
  const int b    = blockIdx.x;

  // ---- init: v[b] -> LDS (padded rows) ----
  const float* vb = vec + b * 4096;
  for (int idx = tid; idx < 4096; idx += 512)
    ldsV[(idx >> 8) * ST_V + (idx & 255)] = vb[idx];

  // ---- persistent stage1 A fragments ----
  // A1[m=(o2=wave, k2=l16)][kk=(k1,i2)] = core1[k1][o2][i2][k2]
  v2f a1[64];
#pragma unroll
  for (int ks = 0; ks < 64; ++ks) {
    const int kk0 = ks * 4 + half * 2;
    const int k1a = kk0 >> 4, i2a = kk0 & 15;
    const float* p = core1 + (((k1a * 16 + wave) * 16 + i2a) * 16 + l16);
    v2f a;
    a.x = p[0];
    a.y = p[16];  // kk0+1 -> i2a+1 (kk0 even => no k1 carry)
    a1[ks] = a;
  }

  __syncthreads();

  for (int o1 = 0; o1 < 16; ++o1) {
    // ---- stage 0: X[k1][n] = sum_i1 core0[o1][i1][k1] * v[i1][n] ----
    // wave owns N-tile ncol = wave*16; M=16 (k1), K=16 (i1)
    {
      const float* A0 = core0 + o1 * 256;  // [i1][k1]
      const int ncol = wave * 16;
      v8f cA = {}, cB = {};
#pragma unroll
      for (int ks = 0; ks < 4; ks += 2) {
        const int kk0 = ks * 4 + half * 2;        // i1 (even step)
        const int kk1 = (ks + 1) * 4 + half * 2;  // i1 (odd step)
        v2f a0, b0, a1f, b1;
        a0.x = A0[kk0 * 16 + l16];  // core0[o1][i1][k1=l16]
        a0.y = A0[(kk0 + 1) * 16 + l16];
        b0.x = ldsV[kk0 * ST_V + ncol + l16];
        b0.y = ldsV[(kk0 + 1) * ST_V + ncol + l16];
        a1f.x = A0[kk1 * 16 + l16];
        a1f.y = A0[(kk1 + 1) * 16 + l16];
        b1.x = ldsV[kk1 * ST_V + ncol + l16];
        b1.y = ldsV[(kk1 + 1) * ST_V + ncol + l16];
        cA = wmma4(a0, b0, cA);
        cB = wmma4(a1f, b1, cB);
      }
      const v8f c = cA + cB;
#pragma unroll
      for (int j = 0; j < 8; ++j)
        ldsX[(j + half * 8) * ST_X + ncol + l16] = c[j];
    }
    __syncthreads();

    // ---- stage 1: Y[(o2,k2)][i3] = sum_(k1,i2) A1 * X ----
    if (tid < 256) ldsT[tid] = 0.0f;  // clear stage2 tile (pre-bar2)
    {
      v8f cA = {}, cB = {};
#pragma unroll
      for (int ks = 0; ks < 64; ks += 2) {
        const int kk0 = ks * 4 + half * 2;
        const int k1a = kk0 >> 4, i2a = kk0 & 15;
        const int kk1 = (ks + 1) * 4 + half * 2;
        const int k1b = kk1 >> 4, i2b = kk1 & 15;
        v2f b0, b1;
        b0.x = ldsX[k1a * ST_X + i2a * 16 + l16];
        b0.y = ldsX[k1a * ST_X + (i2a + 1) * 16 + l16];  // no carry: kk even
        b1.x = ldsX[k1b * ST_X + i2b * 16 + l16];
        b1.y = ldsX[k1b * ST_X + (i2b + 1) * 16 + l16];
        cA = wmma4(a1[ks], b0, cA);
        cB = wmma4(a1[ks + 1], b1, cB);
      }
      const v8f c = cA + cB;
#pragma unroll
      for (int j = 0; j < 8; ++j)
        ldsY[wave * ST_Y + (j + half * 8) * 18 + l16] = c[j];
    }
    __syncthreads();

    // ---- stage 2: T[o3][o2] = sum_(k2,i3) core2[k2][o3][i3] * Y[o2][k2][i3]
    // k-range split across 16 waves, reduced via ds_add_f32
    {
      v8f cA = {}, cB = {};
#pragma unroll
      for (int s = 0; s < 4; s += 2) {
        const int kk0 = (wave * 4 + s) * 4 + half * 2;
        const int k2a = kk0 >> 4, i3a = kk0 & 15;
        const int kk1 = (wave * 4 + s + 1) * 4 + half * 2;
        const int k2b = kk1 >> 4, i3b = kk1 & 15;
        v2f a0, b0, a1f, b1;
        const float* p0 = core2 + ((k2a * 16 + l16) * 16 + i3a);
        a0.x = p0[0];
        a0.y = p0[1];  // i3a+1, no carry (i3a even <= 14)
        b0.x = ldsY[l16 * ST_Y + k2a * 18 + i3a];
        b0.y = ldsY[l16 * ST_Y + k2a * 18 + i3a + 1];
        const float* p1 = core2 + ((k2b * 16 + l16) * 16 + i3b);
        a1f.x = p1[0];
        a1f.y = p1[1];
        b1.x = ldsY[l16 * ST_Y + k2b * 18 + i3b];
        b1.y = ldsY[l16 * ST_Y + k2b * 18 + i3b + 1];
        cA = wmma4(a0, b0, cA);
        cB = wmma4(a1f, b1, cB);
      }
      const v8f c = cA + cB;
#pragma unroll
      for (int j = 0; j < 8; ++j)
        atomicAdd(&ldsT[(j + half * 8) * 16 + l16], c[j]);
    }
    __syncthreads();

    // ---- coalesced store: out[b][o1][o2][o3], tid = o2*16+o3 ----
    if (tid < 256)
      out[b * 4096 + o1 * 256 + tid] = ldsT[(tid & 15) * 16 + (tid >> 4)];
  }
}

extern "C" void kernel_launch(void* const* d_in, const int* in_sizes, int n_in,
                              void* d_out, int out_size, void* d_ws,
                              size_t ws_size, hipStream_t stream) {
  const float* vec   = (const float*)d_in[0];  // (1024,16,16,16)
  const float* core0 = (const float*)d_in[1];  // (1,16,16,16)
  const float* core1 = (const float*)d_in[2];  // (16,16,16,16)
  const float* core2 = (const float*)d_in[3];  // (16,16,16,1)
  float* outp = (float*)d_out;                 // (1024,16,16,16)
  (void)in_sizes; (void)n_in; (void)out_size; (void)d_ws; (void)ws_size;
  tt_matvec_kernel<<<dim3(1024), dim3(512), 0, stream>>>(vec, core0, core1,
                                                         core2, outp);
}